// GATEncoder_50998441673101
// MI455X (gfx1250) — compile-verified
//
#include <hip/hip_runtime.h>
#include <math.h>

typedef float v2f __attribute__((ext_vector_type(2)));
typedef float v8f __attribute__((ext_vector_type(8)));

#define N_NODES   50000
#define IN_DIM    128
#define HID       64
#define HEADS     4
#define OUT_DIM   128
#define HC        (HEADS * HID)        /* 256 */
#define N_EDGES   800000
#define E_TOT     (N_EDGES + N_NODES)  /* 850000, self-loops appended */
#define NEG_SLOPE 0.2f
#define MT        5                    /* M-tiles per wave: 3125 = 5 * 625 */

// ---------------------------------------------------------------------------
// Utility: fill a float buffer with a constant (zero-init / -inf init).
// ---------------------------------------------------------------------------
__global__ void fill_kernel(float* __restrict__ p, float v, int n) {
    int i = blockIdx.x * blockDim.x + threadIdx.x;
    if (i < n) p[i] = v;
}

// ---------------------------------------------------------------------------
// Pack W[K][N] -> Wp[K/2][N][2] so a B fragment (rows k,k+1 at one column)
// is a single aligned 8-byte load.
// ---------------------------------------------------------------------------
__global__ void pack_w_kernel(const float* __restrict__ W, float* __restrict__ Wp,
                              int N, int K) {
    int i = blockIdx.x * blockDim.x + threadIdx.x;
    if (i >= K * N) return;
    int k = i / N, n = i % N;
    Wp[(size_t)(k >> 1) * 2 * N + 2 * n + (k & 1)] = W[i];
}

// ---------------------------------------------------------------------------
// GEMM  C[M,N] = A[M,K] * Wp(K,N packed)  via V_WMMA_F32_16X16X4_F32.
// grid.x = M/(16*MT) = 625; blockDim.x = 32*(N/64).
// Each wave: MT=5 M-tiles x 4 N-subtiles -> 20 WMMAs per K-step of 4,
// fed by 5 A b64 loads + 4 packed-B b64 loads.
//
// A 16x4 fragment: lane l holds row l%16, K = 2*(l>>4)+j  (contiguous -> b64).
// B 4x16 fragment: lane l holds N = l%16, K = 2*(l>>4)+j  (packed    -> b64).
// C/D 16x16:       VGPR r, lane l -> row r + 8*(l>>4), col l%16.
// ---------------------------------------------------------------------------
__global__ void gemm_wmma_f32(const float* __restrict__ A,
                              const float* __restrict__ Wp,
                              float* __restrict__ C,
                              int N, int K) {
    const int lane = threadIdx.x & 31;
    const int wave = threadIdx.x >> 5;
    const int m0   = blockIdx.x * (16 * MT);
    const int n0   = wave * 64;
    const int half = lane >> 4;          // 0 or 1
    const int ncol = n0 + (lane & 15);

    v8f acc[MT][4] = {};

    const float* __restrict__ aptr[MT];
#pragma unroll
    for (int t = 0; t < MT; ++t)
        aptr[t] = A + (size_t)(m0 + t * 16 + (lane & 15)) * K + 2 * half;

    for (int k = 0; k < K; k += 4) {
        v2f a[MT];
#pragma unroll
        for (int t = 0; t < MT; ++t)
            a[t] = *(const v2f*)(aptr[t] + k);

        // packed row (k/2 + half) holds original rows k+2*half, k+2*half+1
        const float* __restrict__ bp =
            Wp + ((size_t)(k >> 1) + half) * 2 * N + 2 * ncol;

#pragma unroll
        for (int nt = 0; nt < 4; ++nt) {
            v2f b = *(const v2f*)(bp + 32 * nt);
#pragma unroll
            for (int t = 0; t < MT; ++t)
                acc[t][nt] = __builtin_amdgcn_wmma_f32_16x16x4_f32(
                    false, a[t], false, b, (short)0, acc[t][nt], false, false);
        }
    }

#pragma unroll
    for (int t = 0; t < MT; ++t) {
#pragma unroll
        for (int r = 0; r < 8; ++r) {
            size_t row = (size_t)(m0 + t * 16 + r + 8 * half);
            float* __restrict__ crow = C + row * N;
#pragma unroll
            for (int nt = 0; nt < 4; ++nt)
                crow[ncol + nt * 16] = acc[t][nt][r];
        }
    }
}

// ---------------------------------------------------------------------------
// Per-node attention scores: s[n,h] = xp[n,h,:].a_src[h], d likewise.
// One thread per (node, head); float4 loads.
// ---------------------------------------------------------------------------
__global__ void scores_kernel(const float* __restrict__ xp,
                              const float* __restrict__ a_src,
                              const float* __restrict__ a_dst,
                              float* __restrict__ s, float* __restrict__ d,
                              int H, int C) {
    int i = blockIdx.x * blockDim.x + threadIdx.x;
    if (i >= N_NODES * H) return;
    int n = i / H, h = i % H;
    const float4* __restrict__ row = (const float4*)(xp + (size_t)n * H * C + (size_t)h * C);
    const float4* __restrict__ as  = (const float4*)(a_src + h * C);
    const float4* __restrict__ ad  = (const float4*)(a_dst + h * C);
    float ss = 0.f, dd = 0.f;
    for (int c = 0; c < C / 4; ++c) {
        float4 v = row[c], a = as[c], b = ad[c];
        ss = fmaf(v.x, a.x, fmaf(v.y, a.y, fmaf(v.z, a.z, fmaf(v.w, a.w, ss))));
        dd = fmaf(v.x, b.x, fmaf(v.y, b.y, fmaf(v.z, b.z, fmaf(v.w, b.w, dd))));
    }
    s[i] = ss;
    d[i] = dd;
}

// Sign-aware float atomic max (works with init = -1e30f).
__device__ __forceinline__ void atomicMaxF(float* addr, float val) {
    if (val >= 0.f)
        atomicMax((int*)addr, __float_as_int(val));
    else
        atomicMin((unsigned int*)addr, (unsigned int)__float_as_int(val));
}

__device__ __forceinline__ void edge_endpoints(const int* __restrict__ ei, int eid,
                                               int& src, int& dst) {
    if (eid < N_EDGES) {
        src = ei[eid];
        dst = ei[N_EDGES + eid];
    } else {
        src = dst = eid - N_EDGES;   // self-loop
    }
}

// ---------------------------------------------------------------------------
// Edge pass 1: e = leaky_relu(s[src] + d[dst]); segment max into m[dst].
// ---------------------------------------------------------------------------
__global__ void edge_score_kernel(const int* __restrict__ ei,
                                  const float* __restrict__ s,
                                  const float* __restrict__ d,
                                  float* __restrict__ e,
                                  float* __restrict__ m, int H) {
    int i = blockIdx.x * blockDim.x + threadIdx.x;
    if (i >= E_TOT * H) return;
    int eid = i / H, h = i % H;
    int src, dst;
    edge_endpoints(ei, eid, src, dst);
    float v = s[src * H + h] + d[dst * H + h];
    v = (v > 0.f) ? v : NEG_SLOPE * v;
    e[i] = v;
    atomicMaxF(&m[dst * H + h], v);
}

// ---------------------------------------------------------------------------
// Edge pass 2: p = exp(e - m[dst]); segment sum into z[dst]; e <- p.
// ---------------------------------------------------------------------------
__global__ void edge_exp_kernel(const int* __restrict__ ei,
                                const float* __restrict__ m,
                                float* __restrict__ e,
                                float* __restrict__ z, int H) {
    int i = blockIdx.x * blockDim.x + threadIdx.x;
    if (i >= E_TOT * H) return;
    int eid = i / H, h = i % H;
    int src, dst;
    edge_endpoints(ei, eid, src, dst);
    float p = __expf(e[i] - m[dst * H + h]);
    e[i] = p;
    atomicAdd(&z[dst * H + h], p);
}

// ---------------------------------------------------------------------------
// Edge pass 3: out[dst] += xp[src] * alpha. One wave per ~4 edges
// (grid-stride); each lane owns H*C/32 contiguous channels (8 or 4), which
// fall inside one head. float4 gathers + prefetch of next edge's src row.
// ---------------------------------------------------------------------------
__global__ void aggregate_kernel(const int* __restrict__ ei,
                                 const float* __restrict__ xp,
                                 const float* __restrict__ e,
                                 const float* __restrict__ z,
                                 float* __restrict__ out, int H, int C) {
    const int lane = threadIdx.x & 31;
    const int wid  = blockIdx.x * (blockDim.x >> 5) + (threadIdx.x >> 5);
    const int nw   = gridDim.x * (blockDim.x >> 5);
    const int hc   = H * C;
    const int per  = hc >> 5;          // channels per lane (8 or 4)
    const int c0   = lane * per;
    const int h    = c0 / C;

    for (int eid = wid; eid < E_TOT; eid += nw) {
        int src, dst;
        edge_endpoints(ei, eid, src, dst);

        // prefetch next edge's source row while this edge's work drains
        int nxt = eid + nw;
        if (nxt < E_TOT) {
            int ns, nd;
            edge_endpoints(ei, nxt, ns, nd);
            __builtin_prefetch(xp + (size_t)ns * hc + c0, 0, 0);
        }

        float alpha = e[(size_t)eid * H + h] / (z[dst * H + h] + 1e-16f);
        const float4* __restrict__ xr = (const float4*)(xp + (size_t)src * hc + c0);
        float* __restrict__ orow      = out + (size_t)dst * hc + c0;
#pragma unroll
        for (int v = 0; v < 2; ++v) {   // per/4 = 2 or 1
            if (v * 4 >= per) break;
            float4 xv = xr[v];
            atomicAdd(&orow[4 * v + 0], xv.x * alpha);
            atomicAdd(&orow[4 * v + 1], xv.y * alpha);
            atomicAdd(&orow[4 * v + 2], xv.z * alpha);
            atomicAdd(&orow[4 * v + 3], xv.w * alpha);
        }
    }
}

// ---------------------------------------------------------------------------
// out += bias (broadcast over rows), optional ELU.
// ---------------------------------------------------------------------------
__global__ void bias_act_kernel(float* __restrict__ out,
                                const float* __restrict__ bias,
                                int hc, int do_elu) {
    int i = blockIdx.x * blockDim.x + threadIdx.x;
    if (i >= N_NODES * hc) return;
    float v = out[i] + bias[i % hc];
    if (do_elu) v = (v > 0.f) ? v : expm1f(v);
    out[i] = v;
}

// ---------------------------------------------------------------------------
// Host-side: one full GATConv layer.
// ---------------------------------------------------------------------------
static void run_gat_layer(const float* in_feat, const float* W,
                          const float* a_src, const float* a_dst,
                          const float* bias,
                          float* xp, float* out, float* wpack,
                          float* s, float* d, float* m, float* z, float* e,
                          const int* ei, int K, int H, int C, int do_elu,
                          hipStream_t stream) {
    const int hc   = H * C;
    const int nOut = N_NODES * hc;
    const int nH   = N_NODES * H;
    const int nEH  = E_TOT * H;

    // Pack weights, then GEMM: xp = in_feat @ W
    pack_w_kernel<<<(K * hc + 255) / 256, 256, 0, stream>>>(W, wpack, hc, K);
    gemm_wmma_f32<<<N_NODES / (16 * MT), 32 * (hc / 64), 0, stream>>>(in_feat, wpack, xp, hc, K);

    // Init accumulators (stream-ordered after GEMM, so safe even when `out`
    // aliases the GEMM input buffer).
    fill_kernel<<<(nOut + 255) / 256, 256, 0, stream>>>(out, 0.f, nOut);
    fill_kernel<<<(nH + 255) / 256, 256, 0, stream>>>(m, -1e30f, nH);
    fill_kernel<<<(nH + 255) / 256, 256, 0, stream>>>(z, 0.f, nH);

    scores_kernel<<<(nH + 255) / 256, 256, 0, stream>>>(xp, a_src, a_dst, s, d, H, C);
    edge_score_kernel<<<(nEH + 255) / 256, 256, 0, stream>>>(ei, s, d, e, m, H);
    edge_exp_kernel<<<(nEH + 255) / 256, 256, 0, stream>>>(ei, m, e, z, H);
    // ~4 edges per wave, 8 waves per block
    aggregate_kernel<<<(E_TOT + 31) / 32, 256, 0, stream>>>(ei, xp, e, z, out, H, C);
    bias_act_kernel<<<(nOut + 255) / 256, 256, 0, stream>>>(out, bias, hc, do_elu);
}

extern "C" void kernel_launch(void* const* d_in, const int* in_sizes, int n_in,
                              void* d_out, int out_size, void* d_ws, size_t ws_size,
                              hipStream_t stream) {
    const float* x      = (const float*)d_in[0];
    const int*   ei     = (const int*)d_in[1];
    const float* W0     = (const float*)d_in[2];
    const float* a_src0 = (const float*)d_in[3];
    const float* a_dst0 = (const float*)d_in[4];
    const float* b0     = (const float*)d_in[5];
    const float* W1     = (const float*)d_in[6];
    const float* a_src1 = (const float*)d_in[7];
    const float* a_dst1 = (const float*)d_in[8];
    const float* b1     = (const float*)d_in[9];
    const float* W2     = (const float*)d_in[10];
    const float* a_src2 = (const float*)d_in[11];
    const float* a_dst2 = (const float*)d_in[12];
    const float* b2     = (const float*)d_in[13];

    // Workspace carve-up (floats). Total ~= 120 MB.
    float* bufA  = (float*)d_ws;                         // xp buffer    (N*HC)
    float* bufB  = bufA + (size_t)N_NODES * HC;          // h/out buffer (N*HC)
    float* sbuf  = bufB + (size_t)N_NODES * HC;          // (N*HEADS)
    float* dbuf  = sbuf + (size_t)N_NODES * HEADS;
    float* mbuf  = dbuf + (size_t)N_NODES * HEADS;
    float* zbuf  = mbuf + (size_t)N_NODES * HEADS;
    float* ebuf  = zbuf + (size_t)N_NODES * HEADS;       // (E_TOT*HEADS)
    float* wpack = ebuf + (size_t)E_TOT * HEADS;         // (max K*N = 65536)

    float* outp = (float*)d_out;

    // Layer 0: x(50000x128) -> h0(50000x256), concat, ELU
    run_gat_layer(x, W0, a_src0, a_dst0, b0,
                  bufA, bufB, wpack, sbuf, dbuf, mbuf, zbuf, ebuf,
                  ei, IN_DIM, HEADS, HID, /*elu=*/1, stream);

    // Layer 1: h0(50000x256) -> h1(50000x256), concat, ELU
    run_gat_layer(bufB, W1, a_src1, a_dst1, b1,
                  bufA, bufB, wpack, sbuf, dbuf, mbuf, zbuf, ebuf,
                  ei, HC, HEADS, HID, /*elu=*/1, stream);

    // Layer 2: h1(50000x256) -> out(50000x128), single head, mean==identity
    run_gat_layer(bufB, W2, a_src2, a_dst2, b2,
                  bufA, outp, wpack, sbuf, dbuf, mbuf, zbuf, ebuf,
                  ei, HC, 1, OUT_DIM, /*elu=*/0, stream);
}